// LightGCN_35253091565751
// MI455X (gfx1250) — compile-verified
//
#include <hip/hip_runtime.h>

#define N_NODES 100000
#define N_FEAT  256
#define EMB     64
#define N_LAYERS 3
#define N_EDGES 1600000
#define N_LBL   1000000

typedef float v2f __attribute__((ext_vector_type(2)));
typedef float v8f __attribute__((ext_vector_type(8)));

#define W_STRIDE 260  // 256 + 4 pad: per-lane b64 LDS reads hit all 64 banks exactly once

// ---------------- utility: zero a float buffer (float4 grid-stride) ----------------
__global__ __launch_bounds__(256) void zero_kernel(float4* __restrict__ p, int n4) {
    int i = blockIdx.x * blockDim.x + threadIdx.x;
    int stride = gridDim.x * blockDim.x;
    float4 z = {0.f, 0.f, 0.f, 0.f};
    for (; i < n4; i += stride) p[i] = z;
}

// ---------------- degree: atomic histogram over destination nodes ----------------
__global__ __launch_bounds__(256) void deg_kernel(const int* __restrict__ col,
                                                  float* __restrict__ deg, int nE) {
    int e = blockIdx.x * blockDim.x + threadIdx.x;
    if (e < nE) unsafeAtomicAdd(&deg[col[e]], 1.0f);
}

// ---------------- deg -> deg_inv_sqrt (in place) ----------------
__global__ __launch_bounds__(256) void dis_kernel(float* __restrict__ deg, int n) {
    int i = blockIdx.x * blockDim.x + threadIdx.x;
    if (i < n) {
        float d = deg[i];
        deg[i] = (d > 0.0f) ? rsqrtf(fmaxf(d, 1.0f)) : 0.0f;
    }
}

// ---------------- embedding GEMM: x = X @ W^T + b ; out = alpha0 * x ----------------
// One wave computes a 16x64 tile of x via V_WMMA_F32_16X16X4_F32.
// Block = 128 threads = 4 waves = 64 rows. W staged in LDS (stride-260 padding,
// conflict-free ds_load_b64 for the B operand).
__global__ __launch_bounds__(128) void embed_wmma(const float* __restrict__ X,
                                                  const float* __restrict__ W,
                                                  const float* __restrict__ bias,
                                                  const float* __restrict__ alpha,
                                                  float* __restrict__ x,
                                                  float* __restrict__ outb) {
    __shared__ float sW[EMB * W_STRIDE];  // ~65 KB of the 320 KB WGP LDS

    const int tid = threadIdx.x;
    // Cooperative W load: 64 rows x 256 floats, float4 moves (row base 1040B = 16B aligned)
    for (int idx = tid; idx < EMB * (N_FEAT / 4); idx += 128) {
        int r  = idx >> 6;        // /64 float4 per row
        int c4 = idx & 63;
        *(float4*)(sW + r * W_STRIDE + c4 * 4) =
            *(const float4*)(W + (size_t)r * N_FEAT + c4 * 4);
    }
    __syncthreads();

    const int wave  = tid >> 5;
    const int lane  = tid & 31;
    const int lhalf = lane >> 4;   // 0: K={k,k+1}, 1: K={k+2,k+3}
    const int lmod  = lane & 15;
    const int mbase = (blockIdx.x * 4 + wave) * 16;

    int arow = mbase + lmod;
    if (arow >= N_NODES) arow = N_NODES - 1;   // clamp (D row m depends only on A row m)
    const float* __restrict__ Xrow = X + (size_t)arow * N_FEAT;

    v8f acc0 = {}, acc1 = {}, acc2 = {}, acc3 = {};

    for (int k = 0; k < N_FEAT; k += 4) {
        const int kk = k + lhalf * 2;
        v2f a = *(const v2f*)(Xrow + kk);

        const float* w0 = sW + (0 * 16 + lmod) * W_STRIDE + kk;
        const float* w1 = sW + (1 * 16 + lmod) * W_STRIDE + kk;
        const float* w2 = sW + (2 * 16 + lmod) * W_STRIDE + kk;
        const float* w3 = sW + (3 * 16 + lmod) * W_STRIDE + kk;
        v2f b0 = *(const v2f*)w0;
        v2f b1 = *(const v2f*)w1;
        v2f b2 = *(const v2f*)w2;
        v2f b3 = *(const v2f*)w3;

        acc0 = __builtin_amdgcn_wmma_f32_16x16x4_f32(false, a, false, b0, (short)0, acc0, false, false);
        acc1 = __builtin_amdgcn_wmma_f32_16x16x4_f32(false, a, false, b1, (short)0, acc1, false, false);
        acc2 = __builtin_amdgcn_wmma_f32_16x16x4_f32(false, a, false, b2, (short)0, acc2, false, false);
        acc3 = __builtin_amdgcn_wmma_f32_16x16x4_f32(false, a, false, b3, (short)0, acc3, false, false);
    }

    const float a0 = alpha[0];
    v8f accs[4] = {acc0, acc1, acc2, acc3};
#pragma unroll
    for (int nt = 0; nt < 4; ++nt) {
        const int colv = nt * 16 + lmod;
        const float bv = bias[colv];
#pragma unroll
        for (int i = 0; i < 8; ++i) {
            const int row = mbase + i + lhalf * 8;   // C/D layout: vgpr i -> M=i / M=8+i
            if (row < N_NODES) {
                float v = accs[nt][i] + bv;
                x[(size_t)row * EMB + colv]    = v;
                outb[(size_t)row * EMB + colv] = v * a0;
            }
        }
    }
}

// ---------------- LGConv scatter: xn[col] += norm * x[row] ----------------
// 16 threads per edge, each handles 4 features (float4 gather + 4 f32 atomics).
__global__ __launch_bounds__(256) void prop_kernel(const float* __restrict__ x,
                                                   const int* __restrict__ row,
                                                   const int* __restrict__ col,
                                                   const float* __restrict__ dis,
                                                   float* __restrict__ xn) {
    long long idx = (long long)blockIdx.x * blockDim.x + threadIdx.x;
    if (idx >= (long long)N_EDGES * 16) return;
    const int e = (int)(idx >> 4);
    const int c = (int)(idx & 15);
    const int r = row[e];
    const int d = col[e];
    const float nrm = dis[r] * dis[d];
    float4 v = ((const float4*)(x + (size_t)r * EMB))[c];
    float* dst = xn + (size_t)d * EMB + c * 4;
    unsafeAtomicAdd(dst + 0, v.x * nrm);
    unsafeAtomicAdd(dst + 1, v.y * nrm);
    unsafeAtomicAdd(dst + 2, v.z * nrm);
    unsafeAtomicAdd(dst + 3, v.w * nrm);
}

// ---------------- out += alpha[ai] * xn ----------------
__global__ __launch_bounds__(256) void axpy_kernel(float4* __restrict__ outb,
                                                   const float4* __restrict__ xn,
                                                   const float* __restrict__ alpha, int ai,
                                                   int n4) {
    int i = blockIdx.x * blockDim.x + threadIdx.x;
    if (i >= n4) return;
    const float a = alpha[ai];
    float4 o = outb[i];
    float4 v = xn[i];
    o.x += a * v.x; o.y += a * v.y; o.z += a * v.z; o.w += a * v.w;
    outb[i] = o;
}

// ---------------- final gather: res[l] = concat(out[src[l]], out[dst[l]]) ----------------
// 32 threads per label edge: c<16 -> src half, c>=16 -> dst half; float4 each.
__global__ __launch_bounds__(256) void gather_kernel(const float* __restrict__ outb,
                                                     const int* __restrict__ src,
                                                     const int* __restrict__ dst,
                                                     float* __restrict__ res) {
    long long idx = (long long)blockIdx.x * blockDim.x + threadIdx.x;
    if (idx >= (long long)N_LBL * 32) return;
    const int l = (int)(idx >> 5);
    const int c = (int)(idx & 31);
    const int node = (c < 16) ? src[l] : dst[l];
    const int cc = c & 15;
    float4 v = ((const float4*)(outb + (size_t)node * EMB))[cc];
    ((float4*)(res + (size_t)l * 128))[c] = v;
}

extern "C" void kernel_launch(void* const* d_in, const int* in_sizes, int n_in,
                              void* d_out, int out_size, void* d_ws, size_t ws_size,
                              hipStream_t stream) {
    const float* X      = (const float*)d_in[0];
    const float* W      = (const float*)d_in[1];
    const float* bias   = (const float*)d_in[2];
    const float* alpha  = (const float*)d_in[3];
    const int*   eidx   = (const int*)d_in[4];
    const int*   lidx   = (const int*)d_in[5];
    const int* erow = eidx;            // [2,E] row-major
    const int* ecol = eidx + N_EDGES;
    const int* lsrc = lidx;
    const int* ldst = lidx + N_LBL;

    float* ws   = (float*)d_ws;
    float* deg  = ws;                       // N (becomes deg_inv_sqrt)
    float* xb   = ws + 131072;              // N*64
    float* xnb  = xb + (size_t)N_NODES * EMB;
    float* ob   = xnb + (size_t)N_NODES * EMB;
    const int nx4 = N_NODES * EMB / 4;      // 1.6M float4

    // 1) degree -> deg_inv_sqrt
    zero_kernel<<<512, 256, 0, stream>>>((float4*)deg, 131072 / 4);
    deg_kernel<<<(N_EDGES + 255) / 256, 256, 0, stream>>>(ecol, deg, N_EDGES);
    dis_kernel<<<(N_NODES + 255) / 256, 256, 0, stream>>>(deg, N_NODES);

    // 2) embedding GEMM (WMMA f32) + out = alpha0*x
    embed_wmma<<<(N_NODES + 63) / 64, 128, 0, stream>>>(X, W, bias, alpha, xb, ob);

    // 3) three LGConv layers
    float* xcur = xb;
    float* xnxt = xnb;
    for (int layer = 0; layer < N_LAYERS; ++layer) {
        zero_kernel<<<4096, 256, 0, stream>>>((float4*)xnxt, nx4);
        prop_kernel<<<(int)(((long long)N_EDGES * 16 + 255) / 256), 256, 0, stream>>>(
            xcur, erow, ecol, deg, xnxt);
        axpy_kernel<<<(nx4 + 255) / 256, 256, 0, stream>>>(
            (float4*)ob, (const float4*)xnxt, alpha, layer + 1, nx4);
        float* t = xcur; xcur = xnxt; xnxt = t;
    }

    // 4) label-edge gather into d_out [L, 128]
    gather_kernel<<<(int)(((long long)N_LBL * 32 + 255) / 256), 256, 0, stream>>>(
        ob, lsrc, ldst, (float*)d_out);
}